// MultiStageIIRFilter_11553462026860
// MI455X (gfx1250) — compile-verified
//
#include <hip/hip_runtime.h>
#include <math.h>
#include <stdint.h>

// Problem constants (from reference)
#define B_      8
#define T_      8000
#define FP_     80
#define NF_     100
#define NTAP    25      // nonzero cepstral coeffs m = 0..24 (rest of CEP pad is zero)
#define PADE_   5
#define BLK     160     // pipeline block = exactly 2 frames
#define NB      (T_ / BLK)
#define THREADS 256

typedef __attribute__((ext_vector_type(2))) float v2f;
typedef __attribute__((ext_vector_type(8))) float v8f;

// ---- LDS layout (floats) ----
#define MC_STR   27                       // row stride; cols 25,26 are hard zeros (K pad)
#define MC_ROWS  101                      // frame 100 duplicates frame 99 (linear_intpl edge)
#define MC_OFF   0
#define MC_PAD   2728                     // 101*27 = 2727, padded so A_OFF+32 is 16B-aligned
#define BUF_SZ   (32 + T_)                // 32-float zero prefix for negative-lag reads
#define A_OFF    (MC_OFF + MC_PAD)
#define Bb_OFF   (A_OFF + BUF_SZ)
#define C_OFF    (Bb_OFF + BUF_SZ)
#define RING_OFF (C_OFF + BUF_SZ)
#define RING_SZ  (5 * 2 * BLK * 2)        // 5 rings x double-buffer x BLK complex
#define LDS_FLOATS (RING_OFF + RING_SZ)   // 30024 floats = ~117 KB

__device__ __forceinline__ float bperm_f(int srcLane, float v) {
  return __int_as_float(
      __builtin_amdgcn_ds_bpermute((srcLane & 31) << 2, __float_as_int(v)));
}

__device__ __forceinline__ float wave_sum(float v) {
#pragma unroll
  for (int off = 16; off > 0; off >>= 1)
    v += bperm_f((int)(threadIdx.x & 31) ^ off, v);
  return v;
}

__device__ __forceinline__ float mc_interp(const float* smc, int m, int t) {
  int f = t / FP_;
  float r = (float)(t - f * FP_) * (1.0f / (float)FP_);
  float a = smc[f * MC_STR + m];
  float b = smc[(f + 1) * MC_STR + m];
  return a + r * (b - a);
}

__global__ void __launch_bounds__(THREADS, 1)
mlsa_pipeline_kernel(const float* __restrict__ x, const float* __restrict__ mc,
                     const float* __restrict__ wts, const float* __restrict__ a1,
                     const float* __restrict__ r1, const float* __restrict__ r2,
                     float* __restrict__ out) {
  extern __shared__ float smem[];
  const int tid  = threadIdx.x;
  const int lane = tid & 31;
  const int wid  = tid >> 5;
  const int b    = blockIdx.x;

  const float* xg  = x + b * T_;
  const float* mcg = mc + b * NF_ * NTAP;
  float* smc = smem + MC_OFF;

  __builtin_prefetch(mcg, 0, 1);   // global_prefetch_b8

  // ---------------- Phase 0: stage inputs in LDS ----------------
  // x -> LDS via CDNA5 async DMA (16B per lane, ASYNCcnt-tracked, no VGPR bounce)
  for (int k = tid; k < T_ / 4; k += THREADS) {
    const float* ga = xg + 4 * k;
    unsigned la = (unsigned)(uintptr_t)(smem + A_OFF + 32 + 4 * k);
    asm volatile("global_load_async_to_lds_b128 %0, %1, off"
                 :: "v"(la), "v"(ga) : "memory");
  }
  if (tid < 32) { smem[A_OFF + tid] = 0.f; smem[Bb_OFF + tid] = 0.f; }
  for (int i = tid; i < NF_ * NTAP; i += THREADS) {
    int f = i / NTAP, m = i - f * NTAP;
    smc[f * MC_STR + m] = mcg[i];
  }
  if (tid < NTAP)    smc[100 * MC_STR + tid] = mcg[99 * NTAP + tid]; // frame 100 = frame 99
  if (tid < MC_ROWS) { smc[tid * MC_STR + 25] = 0.f; smc[tid * MC_STR + 26] = 0.f; }
  asm volatile("s_wait_asynccnt 0x0" ::: "memory");
  __syncthreads();

  // Pade weights / a-coeffs (uniform scalar loads)
  float w_[PADE_ + 1], a_[PADE_ + 1];
#pragma unroll
  for (int i = 0; i <= PADE_; ++i) {
    w_[i] = wts[i];
    a_[i] = fminf(fmaxf(a1[i], 0.1f), 10.0f);
  }
  a_[0] = 1.0f;

  // ---------------- FIR stage 1 (lag-1 cascade, closed form per t) ----------------
  for (int t = tid; t < T_; t += THREADS) {
    float acc = smem[A_OFF + 32 + t];
    float prodc = 1.0f, Wa = 1.0f;
#pragma unroll
    for (int i = 1; i <= PADE_; ++i) {
      Wa *= w_[i];
      int u = t - (i - 1); if (u < 0) u = 0;
      prodc *= mc_interp(smc, 1, u);
      acc += a_[i] * Wa * prodc * smem[A_OFF + 32 + t - i]; // zero-prefix handles t<i
    }
    smem[Bb_OFF + 32 + t] = acc;   // xs_0
    smem[C_OFF + 32 + t]  = acc;   // y2 accumulator, a2c[0] = 1
  }
  __syncthreads();

  // ---------------- FIR stage 2 cascade via fp32 WMMA ----------------
  // xs_i[t] = w_i * sum_{m=2..25} xs_{i-1}[t-m] * cI[t][m]
  //         = w_i * ((1-r)*S_f[t] + r*S_{f+1}[t]),  S = window matmul per frame.
  int Pof = Bb_OFF, Qof = A_OFF;
  for (int stage = 1; stage <= PADE_; ++stage) {
    float wi = w_[stage], ai = a_[stage];
    for (int tile = wid; tile < NF_ * 5; tile += 8) {   // 5 row-tiles of 16 per frame
      int f    = tile / 5;
      int p0   = (tile - f * 5) * 16;
      int t0   = f * FP_ + p0;
      int row  = lane & 15;
      int koff = (lane < 16) ? 0 : 2;   // A: lanes 0-15 hold K={0,1}, 16-31 hold K={2,3}
      int n    = lane & 15;             // B column
      // branch-free B addressing: cols 0/1 read frame f / f+1 rows, others the zero cols
      int brow = ((n == 1) ? (f + 1) : f) * MC_STR;
      bool ncoef = (n <= 1);
      v8f dacc = {0.f, 0.f, 0.f, 0.f, 0.f, 0.f, 0.f, 0.f};
#pragma unroll
      for (int ch = 0; ch < 6; ++ch) {  // K = 24 in 6 chunks of 4 (m = 2..25, m=25 -> 0)
        int m0 = 4 * ch + koff + 2;
        int c0 = ncoef ? m0 : 25;       // lanes n>=2 hit the zero columns 25/26
        v2f afrag, bfrag;
        afrag.x = smem[Pof + 32 + t0 + row - m0];
        afrag.y = smem[Pof + 32 + t0 + row - m0 - 1];
        bfrag.x = smc[brow + c0];
        bfrag.y = smc[brow + c0 + 1];
        dacc = __builtin_amdgcn_wmma_f32_16x16x4_f32(false, afrag, false, bfrag,
                                                     (short)0, dacc, false, false);
      }
      // Extract columns 0 (frame f) and 1 (frame f+1): lanes {0,16} / {1,17}.
      float Sf = 0.f, Sf1 = 0.f;
#pragma unroll
      for (int j = 0; j < 8; ++j) {
        float v0  = bperm_f(0,  dacc[j]);
        float v1  = bperm_f(1,  dacc[j]);
        float v16 = bperm_f(16, dacc[j]);
        float v17 = bperm_f(17, dacc[j]);
        if (lane == j)     { Sf = v0;  Sf1 = v1;  }
        if (lane == j + 8) { Sf = v16; Sf1 = v17; }
      }
      if (lane < 16) {
        int t   = t0 + lane;
        float r = (float)(p0 + lane) * (1.0f / (float)FP_);
        float xs = wi * (Sf + r * (Sf1 - Sf));
        smem[Qof + 32 + t] = xs;
        smem[C_OFF + 32 + t] += ai * xs;
      }
    }
    __syncthreads();
    int tmpo = Pof; Pof = Qof; Qof = tmpo;
  }

  // ---------------- IIR wavefront pipeline: 6 stages across 6 waves ----------------
  // stage 0 (wave 0): five 1st-order complex sections, p1[i]*cI[t][1]
  // stages 1..5 (waves 1..5): order-25 complex section, lags 2..25, p2[s]*cI[t][k]
  float hre = 0.f, him = 0.f;                       // lag history: lane k holds y[t-1-k]
  float z_re[PADE_] = {0.f, 0.f, 0.f, 0.f, 0.f};
  float z_im[PADE_] = {0.f, 0.f, 0.f, 0.f, 0.f};
  float p2re = 0.f, p2im = 0.f;
  if (wid >= 1 && wid <= 5) {
    float zr = r2[2 * (wid - 1)], zi = r2[2 * (wid - 1) + 1];
    float d = zr * zr + zi * zi;
    p2re = zr / d; p2im = -zi / d;                  // 1/root
  }
  float p1re[PADE_], p1im[PADE_];
#pragma unroll
  for (int i = 0; i < PADE_; ++i) {
    float zr = r1[2 * i], zi = r1[2 * i + 1];
    float d = zr * zr + zi * zi;
    p1re[i] = zr / d; p1im[i] = -zi / d;
  }
  const int  kk  = (lane < 27) ? lane : 26;         // clamped coeff read (cols 25/26 = 0)
  const bool act = (lane >= 1 && lane <= 24);       // taps k = 1..24 (lags 2..25)

  for (int q = 0; q < NB + PADE_; ++q) {
    if (wid == 0) {
      int j = q;
      if (j < NB && lane == 0) {
        int t0 = j * BLK;
        int rout = RING_OFF + ((0 * 2 + (j & 1)) * BLK) * 2;
        for (int p = 0; p < BLK; ++p) {
          int t = t0 + p;
          float ci1 = mc_interp(smc, 1, t);
          float cr = smem[C_OFF + 32 + t], cim = 0.f;
#pragma unroll
          for (int s = 0; s < PADE_; ++s) {
            float are = p1re[s] * ci1, aim = p1im[s] * ci1;
            float nr = cr  - (are * z_re[s] - aim * z_im[s]);
            float ni = cim - (are * z_im[s] + aim * z_re[s]);
            z_re[s] = nr; z_im[s] = ni; cr = nr; cim = ni;
          }
          smem[rout + 2 * p] = cr; smem[rout + 2 * p + 1] = cim;
        }
      }
    } else if (wid <= 5) {
      int j = q - wid;
      if (j >= 0 && j < NB) {
        int t0 = j * BLK;
        int rin  = RING_OFF + (((wid - 1) * 2 + (j & 1)) * BLK) * 2;
        int rout = RING_OFF + (((wid < 5 ? wid : 0) * 2 + (j & 1)) * BLK) * 2;
        for (int half = 0; half < 2; ++half) {       // BLK = exactly 2 frames
          int f = j * 2 + half;
          float m0v = smc[f * MC_STR + kk];
          float m1v = smc[(f + 1) * MC_STR + kk];
          for (int p = 0; p < FP_; ++p) {
            int tl = half * FP_ + p;
            float r  = (float)p * (1.0f / (float)FP_);
            float ci = m0v + r * (m1v - m0v);        // cI[t][k]
            float cr  = act ? ci * hre : 0.f;
            float cii = act ? ci * him : 0.f;
            cr  = wave_sum(cr);                      // complex tap-sum (all lanes get it)
            cii = wave_sum(cii);
            float xr = smem[rin + 2 * tl];
            float xi = smem[rin + 2 * tl + 1];
            float yr = xr - (p2re * cr  - p2im * cii);
            float yi = xi - (p2re * cii + p2im * cr);
            float sr = bperm_f(lane - 1, hre);       // shift history down the lanes
            float si = bperm_f(lane - 1, him);
            hre = (lane == 0) ? yr : sr;
            him = (lane == 0) ? yi : si;
            if (lane == 0) {
              if (wid < 5) { smem[rout + 2 * tl] = yr; smem[rout + 2 * tl + 1] = yi; }
              else         { smem[A_OFF + 32 + t0 + tl] = yr; }  // final real part
            }
          }
        }
      }
    }
    __syncthreads();
  }

  // ---------------- Final gain K = exp(interp(mc[...,0])) ----------------
  for (int t = tid; t < T_; t += THREADS) {
    float c0v = mc_interp(smc, 0, t);
    out[b * T_ + t] = smem[A_OFF + 32 + t] * __expf(c0v);
  }
}

extern "C" void kernel_launch(void* const* d_in, const int* in_sizes, int n_in,
                              void* d_out, int out_size, void* d_ws, size_t ws_size,
                              hipStream_t stream) {
  (void)in_sizes; (void)n_in; (void)out_size; (void)d_ws; (void)ws_size;
  const float* x   = (const float*)d_in[0];
  const float* mc  = (const float*)d_in[1];
  const float* wts = (const float*)d_in[2];
  const float* a1  = (const float*)d_in[3];
  const float* r1  = (const float*)d_in[4];   // complex64 roots1 as (re,im) pairs
  const float* r2  = (const float*)d_in[5];   // complex64 roots2 as (re,im) pairs
  float* out = (float*)d_out;
  size_t shbytes = (size_t)LDS_FLOATS * sizeof(float);
  hipLaunchKernelGGL(mlsa_pipeline_kernel, dim3(B_), dim3(THREADS), shbytes, stream,
                     x, mc, wts, a1, r1, r2, out);
}